// GNN_30416958390395
// MI455X (gfx1250) — compile-verified
//
#include <hip/hip_runtime.h>

// ---- problem constants (from reference) ----
#define N_NODES  40000
#define N_EDGES  640000
#define N_GRAPHS 256
#define D        128            // D_IN and all hidden dims
#define D_CAT    512
#define BN_EPS   1e-5f
#define M_TILES  (N_NODES / 16) // 2500, exact
#define MT       2              // M-tiles per wave (B-fragment reuse)
#define TASKS    (M_TILES / MT) // 1250, exact

typedef __attribute__((ext_vector_type(2))) float v2f;
typedef __attribute__((ext_vector_type(8))) float v8f;

// ------------------------------------------------------------------
// count nodes per graph
__global__ void gin_count(const long long* __restrict__ batch, int* __restrict__ counts) {
    int i = blockIdx.x * blockDim.x + threadIdx.x;
    if (i < N_NODES) atomicAdd(&counts[(int)batch[i]], 1);
}

// pooled sums: pools[g][colOff + f] += h[node][f]
__global__ void gin_pool_add(const float* __restrict__ h, const long long* __restrict__ batch,
                             float* __restrict__ pools, int colOff) {
    int tid = blockIdx.x * blockDim.x + threadIdx.x;
    if (tid >= N_NODES * D) return;
    int node = tid >> 7;
    int f    = tid & (D - 1);
    int g    = (int)batch[node];
    atomicAdd(&pools[(size_t)g * D_CAT + colOff + f], h[tid]);
}

// agg[dst] += h[src], 4 features per thread (agg pre-initialized to h)
__global__ void gin_scatter(const float* __restrict__ h,
                            const long long* __restrict__ src,
                            const long long* __restrict__ dst,
                            float* __restrict__ agg) {
    int tid = blockIdx.x * blockDim.x + threadIdx.x;
    int e = tid >> 5;               // 32 threads per edge
    if (e >= N_EDGES) return;
    int f = (tid & 31) << 2;        // 4 floats each
    int s = (int)src[e];
    int d = (int)dst[e];
    const float4 v = *(const float4*)(h + (size_t)s * D + f);
    float* p = agg + (size_t)d * D + f;
    atomicAdd(p + 0, v.x);
    atomicAdd(p + 1, v.y);
    atomicAdd(p + 2, v.z);
    atomicAdd(p + 3, v.w);
}

// ------------------------------------------------------------------
// X @ W + b -> ReLU -> BN, via V_WMMA_F32_16X16X4_F32.
// Each wave computes MT (=2) 16x128 output strips (8 N-tiles each):
// B fragments are loaded once per K-step and reused for both M tiles,
// giving 16 WMMAs per 8 LDS pair-loads. 8 waves / block.
// W (128x128) staged in LDS row-major: B-frag reads are conflict-free
// (lanes read consecutive columns of one K-row).
__global__ void __launch_bounds__(256)
gin_mlp(const float* __restrict__ X,
        const float* __restrict__ W, const float* __restrict__ bias,
        const float* __restrict__ gam, const float* __restrict__ bet,
        const float* __restrict__ rm,  const float* __restrict__ rv,
        float* __restrict__ out) {
    __shared__ float sW[D * D];       // 64 KB (<< 320 KB/WGP)
    for (int i = threadIdx.x; i < D * D; i += 256) sW[i] = W[i];
    __syncthreads();

    const int wave = threadIdx.x >> 5;         // uniform per wave
    const int lane = threadIdx.x & 31;
    const int task = blockIdx.x * 8 + wave;    // MT consecutive 16-row tiles
    if (task >= TASKS) return;                 // wave-uniform: EXEC stays all-1s
    const int tile0 = task * MT;

    const int m  = lane & 15;                  // A row within tile
    const int n  = lane & 15;                  // B/C col within 16-tile
    const int kk = (lane >> 4) << 1;           // K sub-offset {0,2}

    const float* xrow[MT];
#pragma unroll
    for (int t = 0; t < MT; ++t)
        xrow[t] = X + ((size_t)(tile0 + t) * 16 + m) * D;

    v8f acc[MT][8];
    const v8f zero = {0.f,0.f,0.f,0.f,0.f,0.f,0.f,0.f};
#pragma unroll
    for (int t = 0; t < MT; ++t)
#pragma unroll
        for (int nt = 0; nt < 8; ++nt) acc[t][nt] = zero;

    for (int k0 = 0; k0 < D; k0 += 4) {
        v2f a[MT];
#pragma unroll
        for (int t = 0; t < MT; ++t) {         // A: K = k0+kk, k0+kk+1
            a[t].x = xrow[t][k0 + kk + 0];
            a[t].y = xrow[t][k0 + kk + 1];
        }
#pragma unroll
        for (int nt = 0; nt < 8; ++nt) {
            int c = nt * 16 + n;
            v2f b;                              // B: same K mapping, col c
            b.x = sW[(k0 + kk + 0) * D + c];
            b.y = sW[(k0 + kk + 1) * D + c];
#pragma unroll
            for (int t = 0; t < MT; ++t)
                acc[t][nt] = __builtin_amdgcn_wmma_f32_16x16x4_f32(
                    false, a[t], false, b, (short)0, acc[t][nt], false, false);
        }
    }

    // epilogue: bias -> ReLU -> BN -> store
    const int rbase = (lane >> 4) << 3;        // row offset {0,8}
#pragma unroll
    for (int nt = 0; nt < 8; ++nt) {
        int c = nt * 16 + n;
        float sc = gam[c] * rsqrtf(rv[c] + BN_EPS);
        float sh = bet[c] - rm[c] * sc;
        float bi = bias[c];
#pragma unroll
        for (int t = 0; t < MT; ++t) {
#pragma unroll
            for (int r = 0; r < 8; ++r) {
                float v = acc[t][nt][r] + bi;
                v = fmaxf(v, 0.0f);
                v = v * sc + sh;
                out[((size_t)(tile0 + t) * 16 + rbase + r) * D + c] = v;
            }
        }
    }
}

// ------------------------------------------------------------------
// head: mean -> BN -> linear(512->1). One thread per graph.
__global__ void gin_head(const float* __restrict__ pools, const int* __restrict__ counts,
                         const float* __restrict__ g, const float* __restrict__ b,
                         const float* __restrict__ rm, const float* __restrict__ rv,
                         const float* __restrict__ Wl, const float* __restrict__ bl,
                         float* __restrict__ out) {
    int gi = blockIdx.x * blockDim.x + threadIdx.x;
    if (gi >= N_GRAPHS) return;
    float inv = 1.0f / fmaxf((float)counts[gi], 1.0f);
    float acc = 0.0f;
    for (int c = 0; c < D_CAT; ++c) {
        float v = pools[(size_t)gi * D_CAT + c] * inv;
        v = (v - rm[c]) * rsqrtf(rv[c] + BN_EPS) * g[c] + b[c];
        acc += v * Wl[c];
    }
    out[gi] = acc + bl[0];
}

// ------------------------------------------------------------------
extern "C" void kernel_launch(void* const* d_in, const int* in_sizes, int n_in,
                              void* d_out, int out_size, void* d_ws, size_t ws_size,
                              hipStream_t stream) {
    (void)in_sizes; (void)n_in; (void)out_size; (void)ws_size;

    // inputs, flattened in setup_inputs() insertion order
    const float*     x     = (const float*)d_in[0];
    const long long* eidx  = (const long long*)d_in[1];
    const long long* batch = (const long long*)d_in[2];
    const long long* src   = eidx;
    const long long* dst   = eidx + N_EDGES;
    // params: layers (W1,b1,g1,be1,W2,b2,g2,be2) x3, then bn_g, bn_b, Wl, bl
    const float* W1[3], *b1[3], *g1[3], *be1[3], *W2[3], *b2[3], *g2[3], *be2[3];
    for (int l = 0; l < 3; ++l) {
        int base = 3 + l * 8;
        W1[l]  = (const float*)d_in[base + 0];
        b1[l]  = (const float*)d_in[base + 1];
        g1[l]  = (const float*)d_in[base + 2];
        be1[l] = (const float*)d_in[base + 3];
        W2[l]  = (const float*)d_in[base + 4];
        b2[l]  = (const float*)d_in[base + 5];
        g2[l]  = (const float*)d_in[base + 6];
        be2[l] = (const float*)d_in[base + 7];
    }
    const float* bn_g = (const float*)d_in[27];
    const float* bn_b = (const float*)d_in[28];
    const float* Wl   = (const float*)d_in[29];
    const float* bl   = (const float*)d_in[30];
    // stats: layers (rm1,rv1,rm2,rv2) x3, then bn_rm, bn_rv
    const float* rm1[3], *rv1[3], *rm2[3], *rv2[3];
    for (int l = 0; l < 3; ++l) {
        int base = 31 + l * 4;
        rm1[l] = (const float*)d_in[base + 0];
        rv1[l] = (const float*)d_in[base + 1];
        rm2[l] = (const float*)d_in[base + 2];
        rv2[l] = (const float*)d_in[base + 3];
    }
    const float* bn_rm = (const float*)d_in[43];
    const float* bn_rv = (const float*)d_in[44];

    // workspace layout (~62 MB)
    const size_t NF = (size_t)N_NODES * D;
    float* agg   = (float*)d_ws;
    float* z1    = agg + NF;
    float* hbuf  = z1 + NF;
    float* pools = hbuf + NF;                    // [256][512]
    int*   cnts  = (int*)(pools + (size_t)N_GRAPHS * D_CAT);

    hipMemsetAsync(pools, 0, (size_t)N_GRAPHS * D_CAT * sizeof(float), stream);
    hipMemsetAsync(cnts,  0, (size_t)N_GRAPHS * sizeof(int), stream);

    gin_count<<<(N_NODES + 255) / 256, 256, 0, stream>>>(batch, cnts);
    gin_pool_add<<<(N_NODES * D + 255) / 256, 256, 0, stream>>>(x, batch, pools, 0);

    const int gemmBlocks = (TASKS + 7) / 8;      // 157
    const float* hcur = x;
    for (int l = 0; l < 3; ++l) {
        // agg := h, then agg[dst] += h[src]  ->  agg = (1+eps)*h + sum_neighbors
        hipMemcpyAsync(agg, hcur, NF * sizeof(float), hipMemcpyDeviceToDevice, stream);
        gin_scatter<<<(N_EDGES * 32 + 255) / 256, 256, 0, stream>>>(hcur, src, dst, agg);
        gin_mlp<<<gemmBlocks, 256, 0, stream>>>(agg, W1[l], b1[l], g1[l], be1[l],
                                                rm1[l], rv1[l], z1);
        gin_mlp<<<gemmBlocks, 256, 0, stream>>>(z1, W2[l], b2[l], g2[l], be2[l],
                                                rm2[l], rv2[l], hbuf);
        gin_pool_add<<<(N_NODES * D + 255) / 256, 256, 0, stream>>>(hbuf, batch, pools,
                                                                    D * (l + 1));
        hcur = hbuf;
    }

    gin_head<<<1, 256, 0, stream>>>(pools, cnts, bn_g, bn_b, bn_rm, bn_rv, Wl, bl,
                                    (float*)d_out);
}